// RegularizedKANActivation_6588479832576
// MI455X (gfx1250) — compile-verified
//
#include <hip/hip_runtime.h>
#include <hip/hip_bf16.h>
#include <math.h>
#include <stdint.h>

// RegularizedKANActivation for MI455X (gfx1250, wave32).
// x:[B,512] f32, control_points:[512,9] f32, reg_weights:[512] f32 -> out:[B,512] f32.
//
// Roofline: ~64MB traffic => ~2.8us @ 23.3 TB/s. The naive Cox-de Boor cascade is
// ~55 VALU/elem (VALU-bound). Instead: on each knot interval the whole spline
// (incl. the reference's stale-column quirks) is exactly a cubic in t per d.
// Kernel 1 fits per-(d,interval) cubic coefficients by sampling the faithful
// cascade (exact divided differences), folding in reg_weights. Kernel 2 does
// v_tanh_f32 -> interval (exact half-open compares) -> one conflict-free
// ds_load_b128 -> 3-FMA Horner -> clamp.
// CDNA5 paths: async global->LDS staging + s_wait_asynccnt, v_tanh_f32.

#define KAN_D    512
#define KAN_NCP  9
#define KAN_NI   6       // 5 real intervals + one all-zero plane for t >= 1.0
#define KAN_TPB  512     // main kernel: 16 wave32 waves; lane owns column d = tid
#define KAN_ROWS 32      // rows of x per block -> 512 blocks

#define AS1 __attribute__((address_space(1)))
#define AS3 __attribute__((address_space(3)))

// Matches the builtin's parameter type: 'vector_size(16) int AS1/AS3 *'
typedef int kan_v4i __attribute__((vector_size(16)));
typedef AS1 kan_v4i* kan_gv4i_p;
typedef AS3 kan_v4i* kan_lv4i_p;

#if defined(__gfx1250__) && __has_builtin(__builtin_amdgcn_global_load_async_to_lds_b128) && __has_builtin(__builtin_amdgcn_s_wait_asynccnt)
#define KAN_ASYNC_LDS 1
#else
#define KAN_ASYNC_LDS 0
#endif

__device__ __forceinline__ float kan_tanh(float v) {
#if defined(__gfx1250__) && __has_builtin(__builtin_amdgcn_tanhf)
  return __builtin_amdgcn_tanhf(v);   // v_tanh_f32 (gfx1250 TRANS op)
#elif defined(__gfx1250__)
  // Guaranteed hardware tanh; trailing v_nop satisfies the TRANS-op result
  // hazard (ISA: 1 independent op or V_NOP before the output is consumed).
  float r;
  asm("v_tanh_f32 %0, %1\n\tv_nop" : "=v"(r) : "v"(v));
  return r;
#else
  return tanhf(v);
#endif
}

// ---------------- Kernel 2 (hot): steady-state evaluation ----------------
// Lane owns column d = tid; LDS layout [i][d][4] makes the per-lane b128 reads
// bank-conflict-free (interval stride = 2048 dwords = multiple of 64 banks;
// d stride = 4 dwords tiles the banks across lanes).
__global__ __launch_bounds__(KAN_TPB) void kan_main(
    const float* __restrict__ x, const float* __restrict__ coef,
    float* __restrict__ out, int B) {
  __shared__ float s_coef[KAN_NI * KAN_D * 4];   // 12288 f32 = 48 KiB

  const int tid = threadIdx.x;

#if KAN_ASYNC_LDS
  {
    #pragma unroll
    for (int k = 0; k < KAN_NI; ++k) {           // 3072 float4 / 512 thr = 6
      const int f4 = tid + k * KAN_TPB;
      __builtin_amdgcn_global_load_async_to_lds_b128(
          (kan_gv4i_p)(coef + 4 * f4),
          (kan_lv4i_p)((AS3 float*)s_coef + 4 * f4),
          0, 0);
    }
    __builtin_amdgcn_s_wait_asynccnt(0);
  }
#else
  for (int i = tid; i < KAN_NI * KAN_D * 4; i += KAN_TPB) s_coef[i] = coef[i];
#endif
  __syncthreads();

  const float4* s_cf4 = reinterpret_cast<const float4*>(s_coef);
  const int d = tid;
  const int row0 = blockIdx.x * KAN_ROWS;

  #pragma unroll 4
  for (int r = 0; r < KAN_ROWS; ++r) {
    const int b = row0 + r;
    if (b >= B) return;
    const size_t off = (size_t)b * KAN_D + d;
    const float t = kan_tanh(x[off]);
    // Exact half-open interval index (same fp32 knots/compares as reference).
    int i = 0;
    i += (t >= -0.6f);
    i += (t >= -0.2f);
    i += (t >=  0.2f);
    i += (t >=  0.6f);
    i += (t >=  1.0f);                     // -> zero plane: spline == 0
    const float u = t - fmaf(0.4f, (float)i, -0.8f);
    const float4 cf = s_cf4[(i << 9) + d]; // ds_load_b128, conflict-free
    const float p = fmaf(fmaf(fmaf(cf.w, u, cf.z), u, cf.y), u, cf.x);
    out[off] = fminf(fmaxf(p, -10.0f), 10.0f);
  }
}

// ---------------- Kernel 1 (tiny): coefficient fitting ----------------
// Faithful Cox-de Boor cascade (order 3, clamped fp32 knots
// {-1,-1,-1,-1,-0.6,-0.2,0.2,0.6,1,1,1,1}), reproducing the reference's
// in-place update: final columns are {b3[0..4], b2[5], b1[6], b0[7]}.
// Returns sum_j basis_j(t) * cp[j]  (no reg_weight, no clamp).
__device__ __forceinline__ float kan_spline(float t, const float* __restrict__ cp) {
  const float c3 = (t >= -1.0f && t < -0.6f) ? 1.0f : 0.0f;
  const float c4 = (t >= -0.6f && t < -0.2f) ? 1.0f : 0.0f;
  const float c5 = (t >= -0.2f && t <  0.2f) ? 1.0f : 0.0f;
  const float c6 = (t >=  0.2f && t <  0.6f) ? 1.0f : 0.0f;
  const float c7 = (t >=  0.6f && t <  1.0f) ? 1.0f : 0.0f;

  const float r04 = 2.5f, r08 = 1.25f, r12 = 0.83333333333333333f;
  const float tp1  = t + 1.0f;
  const float tm06 = -0.6f - t;
  const float tn02 = -0.2f - t;
  const float tp06 = t + 0.6f;
  const float q02  = 0.2f - t;
  const float tp02 = t + 0.2f;
  const float q06  = 0.6f - t;
  const float tm02 = t - 0.2f;
  const float q1   = 1.0f - t;

  const float b12 = tm06 * r04 * c3;
  const float b13 = tp1  * r04 * c3 + tn02 * r04 * c4;
  const float b14 = tp06 * r04 * c4 + q02  * r04 * c5;
  const float b15 = tp02 * r04 * c5 + q06  * r04 * c6;
  const float b16 = tm02 * r04 * c6 + q1   * r04 * c7;

  const float b21 = tm06 * r04 * b12;
  const float b22 = tp1  * r04 * b12 + tn02 * r08 * b13;
  const float b23 = tp1  * r08 * b13 + q02  * r08 * b14;
  const float b24 = tp06 * r08 * b14 + q06  * r08 * b15;
  const float b25 = tp02 * r08 * b15 + q1   * r08 * b16;

  const float b30 = tm06 * r04 * b21;
  const float b31 = tp1  * r04 * b21 + tn02 * r08 * b22;
  const float b32 = tp1  * r08 * b22 + q02  * r12 * b23;
  const float b33 = tp1  * r12 * b23 + q06  * r12 * b24;
  const float b34 = tp06 * r12 * b24 + q1   * r12 * b25;

  return b30 * cp[0] + b31 * cp[1] + b32 * cp[2] + b33 * cp[3]
       + b34 * cp[4] + b25 * cp[5] + b16 * cp[6] + c7  * cp[7];
}

// Fit coef[i][d][0..3] (cubic in u = t - center_i) by sampling the cascade at
// center_i +/- {0.05, 0.15}; exact for cubics (symmetric divided differences).
// rw folded in. Layout: coef[(i*512 + d)*4 + c]; i=5 plane is all zeros.
__global__ __launch_bounds__(256) void kan_precompute(
    const float* __restrict__ cp, const float* __restrict__ rw,
    float* __restrict__ coef) {
  const int idx = blockIdx.x * 256 + threadIdx.x;   // (i*512 + d)
  if (idx >= KAN_D * KAN_NI) return;
  const int d = idx & (KAN_D - 1);
  const int i = idx >> 9;

  float4 cf = make_float4(0.0f, 0.0f, 0.0f, 0.0f);
  if (i < 5) {
    float cpl[8];
    #pragma unroll
    for (int j = 0; j < 8; ++j) cpl[j] = cp[d * KAN_NCP + j];

    const float c  = fmaf(0.4f, (float)i, -0.8f);   // interval center
    const float sm3 = kan_spline(c - 0.15f, cpl);
    const float sm1 = kan_spline(c - 0.05f, cpl);
    const float s1  = kan_spline(c + 0.05f, cpl);
    const float s3  = kan_spline(c + 0.15f, cpl);

    const float e1 = s1 + sm1, e3 = s3 + sm3;
    const float o1 = s1 - sm1, o3 = s3 - sm3;
    // h = 0.05: a0=(9e1-e3)/16, a2=(e3-e1)/(16h^2), a1=(27o1-o3)/(48h), a3=(o3-3o1)/(48h^3)
    const float a0 = (9.0f * e1 - e3) * 0.0625f;
    const float a2 = (e3 - e1) * 25.0f;
    const float a1 = (27.0f * o1 - o3) * 0.41666666666666667f;
    const float a3 = (o3 - 3.0f * o1) * 166.66666666666667f;

    const float w = rw[d];
    cf = make_float4(a0 * w, a1 * w, a2 * w, a3 * w);
  }
  *reinterpret_cast<float4*>(coef + (size_t)idx * 4) = cf;
}

extern "C" void kernel_launch(void* const* d_in, const int* in_sizes, int n_in,
                              void* d_out, int out_size, void* d_ws, size_t ws_size,
                              hipStream_t stream) {
  (void)n_in; (void)out_size; (void)ws_size;
  const float* x  = (const float*)d_in[0];
  const float* cp = (const float*)d_in[1];
  const float* rw = (const float*)d_in[2];
  float* out  = (float*)d_out;
  float* coef = (float*)d_ws;                 // 6*512*4 f32 = 48 KiB scratch

  // 1) fit per-(d,interval) cubics (deterministic, redone every call)
  const int nfit = KAN_D * KAN_NI;            // 3072
  kan_precompute<<<(nfit + 255) / 256, 256, 0, stream>>>(cp, rw, coef);

  // 2) evaluate
  const int B = in_sizes[0] / KAN_D;          // 16384
  const int blocks = (B + KAN_ROWS - 1) / KAN_ROWS;   // 512
  kan_main<<<blocks, KAN_TPB, 0, stream>>>(x, coef, out, B);
}